// DSRLossStateless_57930518889023
// MI455X (gfx1250) — compile-verified
//
#include <hip/hip_runtime.h>

typedef __attribute__((ext_vector_type(2))) float v2f;
typedef __attribute__((ext_vector_type(8))) float v8f;

#define ETA 0.01f
#define EPSF 1e-8f
#define CHUNK 512          // scan chunk: 3907 chunks for B=2M -> ~122 waves
#define PD 4               // prefetch depth (float4 groups in flight)

// ---------------------------------------------------------------------------
// Phase 1: R[t] = dot(W[t,:], X[t,:]).  One 16-timestep tile per wave via
// V_WMMA_F32_16X16X4_F32:  D = Ones(16x16) x P,  P[k][n] = W[t=n][k]*X[t=n][k],
// so every row of D holds the 16 dot products.  A = all ones (layout
// invariant); B lane map (32-bit 4x16): lane l<16 -> (K=0..1, N=l),
// lane l>=16 -> (K=2..3, N=l-16); 4 K-chunks cover 16 assets via C-accum.
// ---------------------------------------------------------------------------
__global__ void __launch_bounds__(256)
rowdot_wmma_kernel(const float* __restrict__ W, const float* __restrict__ X,
                   float* __restrict__ R, int numTiles, int Brows) {
  int wave = blockIdx.x * (blockDim.x >> 5) + (threadIdx.x >> 5);
  int lane = threadIdx.x & 31;
  if (wave >= numTiles) return;              // wave-uniform branch
  int n  = lane & 15;                        // timestep-in-tile (N column)
  int hi = lane >> 4;                        // K half: 0 -> K0..1, 1 -> K2..3
  long row = (long)wave * 16 + n;
  if (row >= Brows) row = Brows - 1;         // per-lane clamp (cndmask)
  const float* wr = W + row * 16 + hi * 2;
  const float* xr = X + row * 16 + hi * 2;

  v8f acc = {};
  v2f ones; ones.x = 1.0f; ones.y = 1.0f;
#pragma unroll
  for (int c = 0; c < 4; ++c) {              // assets 4c..4c+3 as K=0..3
    v2f wv = *(const v2f*)(wr + 4 * c);
    v2f xv = *(const v2f*)(xr + 4 * c);
    v2f p  = wv * xv;
    acc = __builtin_amdgcn_wmma_f32_16x16x4_f32(
        false, ones, false, p, (short)0, acc, false, false);
  }
  long t = (long)wave * 16 + n;
  if (hi == 0 && t < Brows) R[t] = acc[0];   // D[0][n] == R_{t=n}
}

// ---------------------------------------------------------------------------
// Phase 2a: per-chunk EMA sums with zero initial state (chunk carry offsets).
// float4 stream with PD-deep rolling prefetch to keep loads in flight.
// ---------------------------------------------------------------------------
__global__ void chunk_sums_kernel(const float* __restrict__ R,
                                  float* __restrict__ SA, float* __restrict__ SB,
                                  int numChunks, int Brows) {
  int c = blockIdx.x * blockDim.x + threadIdx.x;
  if (c >= numChunks) return;
  long base = (long)c * CHUNK;
  long rem  = (long)Brows - base;
  int  len  = rem < CHUNK ? (int)rem : CHUNK;
  const float*  p  = R + base;
  const float4* p4 = (const float4*)p;
  int nvec = len >> 2;

  float sA = 0.0f, sB = 0.0f;
  float4 buf[PD];
#pragma unroll
  for (int k = 0; k < PD; ++k) {
    float4 z; z.x = z.y = z.z = z.w = 0.0f;
    buf[k] = (k < nvec) ? p4[k] : z;
  }
#pragma unroll 4
  for (int i = 0; i < nvec; ++i) {
    float4 cur = buf[i & (PD - 1)];
    int pf = i + PD;
    if (pf < nvec) buf[i & (PD - 1)] = p4[pf];
    float r;
    r = cur.x; sA = fmaf(ETA, r - sA, sA); sB = fmaf(ETA, fmaf(r, r, -sB), sB);
    r = cur.y; sA = fmaf(ETA, r - sA, sA); sB = fmaf(ETA, fmaf(r, r, -sB), sB);
    r = cur.z; sA = fmaf(ETA, r - sA, sA); sB = fmaf(ETA, fmaf(r, r, -sB), sB);
    r = cur.w; sA = fmaf(ETA, r - sA, sA); sB = fmaf(ETA, fmaf(r, r, -sB), sB);
  }
  for (int i = nvec << 2; i < len; ++i) {
    float r = p[i];
    sA = fmaf(ETA, r - sA, sA);
    sB = fmaf(ETA, fmaf(r, r, -sB), sB);
  }
  SA[c] = sA; SB[c] = sB;
}

// ---------------------------------------------------------------------------
// Phase 2b: exclusive scan of chunk carries.  A_out = d^CHUNK * A_in + S_A.
// Two interleaved FMA chains, ~6 cyc/iter over an L2-resident buffer.
// (Last chunk's carry-out is dead, so only full-chunk decay is needed.)
// ---------------------------------------------------------------------------
__global__ void carry_scan_kernel(const float* __restrict__ SA,
                                  const float* __restrict__ SB,
                                  float* __restrict__ Ain, float* __restrict__ Bin,
                                  int numChunks) {
  if (threadIdx.x != 0 || blockIdx.x != 0) return;
  float decay = __powf(1.0f - ETA, (float)CHUNK);
  float A = 0.0f, Bv = EPSF;
  for (int c = 0; c < numChunks; ++c) {
    Ain[c] = A; Bin[c] = Bv;
    A  = fmaf(decay, A,  SA[c]);
    Bv = fmaf(decay, Bv, SB[c]);
  }
}

// ---------------------------------------------------------------------------
// Phase 2c: replay each chunk with its carry-in, accumulate D_t partials.
// Same pipelined float4 stream as phase 2a.
// ---------------------------------------------------------------------------
__global__ void dsr_kernel(const float* __restrict__ R,
                           const float* __restrict__ Ain, const float* __restrict__ Bin,
                           float* __restrict__ partial, int numChunks, int Brows) {
  int c = blockIdx.x * blockDim.x + threadIdx.x;
  if (c >= numChunks) return;
  long base = (long)c * CHUNK;
  long rem  = (long)Brows - base;
  int  len  = rem < CHUNK ? (int)rem : CHUNK;
  const float*  p  = R + base;
  const float4* p4 = (const float4*)p;
  int nvec = len >> 2;

  float A = Ain[c], Bv = Bin[c], acc = 0.0f;
  float4 buf[PD];
#pragma unroll
  for (int k = 0; k < PD; ++k) {
    float4 z; z.x = z.y = z.z = z.w = 0.0f;
    buf[k] = (k < nvec) ? p4[k] : z;
  }
#pragma unroll 4
  for (int i = 0; i < nvec; ++i) {
    float4 cur = buf[i & (PD - 1)];
    int pf = i + PD;
    if (pf < nvec) buf[i & (PD - 1)] = p4[pf];
#pragma unroll
    for (int e = 0; e < 4; ++e) {
      float r = (e == 0) ? cur.x : (e == 1) ? cur.y : (e == 2) ? cur.z : cur.w;
      float dA = ETA * (r - A);
      float dB = ETA * fmaf(r, r, -Bv);
      float v  = fmaxf(Bv - A * A, EPSF);
      float denom = v * sqrtf(v);                      // v^1.5
      acc += fmaf(Bv, dA, -0.5f * A * dB) / denom;
      A  += dA;
      Bv += dB;
    }
  }
  for (int i = nvec << 2; i < len; ++i) {
    float r = p[i];
    float dA = ETA * (r - A);
    float dB = ETA * fmaf(r, r, -Bv);
    float v  = fmaxf(Bv - A * A, EPSF);
    float denom = v * sqrtf(v);
    acc += fmaf(Bv, dA, -0.5f * A * dB) / denom;
    A  += dA;
    Bv += dB;
  }
  partial[c] = acc;
}

// ---------------------------------------------------------------------------
// Phase 3: deterministic single-block reduction, out = -sum/B.
// ---------------------------------------------------------------------------
__global__ void reduce_kernel(const float* __restrict__ partial, float* __restrict__ out,
                              int numChunks, float invB) {
  __shared__ float smem[256];
  float s = 0.0f;
  for (int i = threadIdx.x; i < numChunks; i += blockDim.x) s += partial[i];
  smem[threadIdx.x] = s;
  __syncthreads();
  for (int off = 128; off > 0; off >>= 1) {
    if ((int)threadIdx.x < off) smem[threadIdx.x] += smem[threadIdx.x + off];
    __syncthreads();
  }
  if (threadIdx.x == 0) out[0] = -smem[0] * invB;
}

// ---------------------------------------------------------------------------
extern "C" void kernel_launch(void* const* d_in, const int* in_sizes, int n_in,
                              void* d_out, int out_size, void* d_ws, size_t ws_size,
                              hipStream_t stream) {
  const float* W = (const float*)d_in[0];   // weights      (B,16) f32
  const float* X = (const float*)d_in[1];   // next_returns (B,16) f32
  int Brows = in_sizes[0] / 16;

  // workspace: R[Brows] | SA | SB | Ain | Bin | partial  (all f32)
  float* R = (float*)d_ws;
  size_t off = (((size_t)Brows * sizeof(float)) + 255) & ~(size_t)255;
  int numChunks = (Brows + CHUNK - 1) / CHUNK;
  float* SA      = (float*)((char*)d_ws + off);
  float* SB      = SA + numChunks;
  float* Ain     = SB + numChunks;
  float* Bin     = Ain + numChunks;
  float* partial = Bin + numChunks;

  int numTiles = (Brows + 15) / 16;
  int blocks1  = (numTiles + 7) / 8;        // 8 waves (tiles) per 256-thr block
  rowdot_wmma_kernel<<<blocks1, 256, 0, stream>>>(W, X, R, numTiles, Brows);

  int blocks2 = (numChunks + 255) / 256;
  chunk_sums_kernel<<<blocks2, 256, 0, stream>>>(R, SA, SB, numChunks, Brows);
  carry_scan_kernel<<<1, 32, 0, stream>>>(SA, SB, Ain, Bin, numChunks);
  dsr_kernel<<<blocks2, 256, 0, stream>>>(R, Ain, Bin, partial, numChunks, Brows);
  reduce_kernel<<<1, 256, 0, stream>>>(partial, (float*)d_out, numChunks,
                                       1.0f / (float)Brows);
}